// GAT_13975823581429
// MI455X (gfx1250) — compile-verified
//
#include <hip/hip_runtime.h>
#include <math.h>
#include <stdint.h>

#define NODES      50000
#define F_IN       256
#define HID        64
#define HEADS      4
#define NCLASS     40
#define NCLASS_P   48                  // padded to a multiple of 16
#define NEDGE      1600000
#define ETOT       (NEDGE + NODES)     // with self loops
#define NEG_SLOPE  0.2f
#define BN_EPS     1e-5f
#define KEY_NEGINF 0x007FFFFFu         // ordered key of -inf

typedef __attribute__((ext_vector_type(2))) float v2f;
typedef __attribute__((ext_vector_type(8))) float v8f;

// ---- monotone float <-> uint key for atomicMax-based segment_max ----
__device__ __forceinline__ unsigned fkey(float f) {
    unsigned b = __float_as_uint(f);
    return (b & 0x80000000u) ? ~b : (b | 0x80000000u);
}
__device__ __forceinline__ float funkey(unsigned k) {
    unsigned b = (k & 0x80000000u) ? (k ^ 0x80000000u) : ~k;
    return __uint_as_float(b);
}
__device__ __forceinline__ float lrelu(float x) { return x > 0.f ? x : NEG_SLOPE * x; }

// async 16B global -> LDS copy (CDNA5 TDM-adjacent path, ASYNCcnt-tracked)
__device__ __forceinline__ void async_g2l_b128(void* lds_ptr, const void* gptr) {
    unsigned lds = (unsigned)(uintptr_t)lds_ptr;   // low 32 bits = LDS offset
    asm volatile("global_load_async_to_lds_b128 %0, %1, off"
                 :: "v"(lds), "v"(gptr) : "memory");
}
__device__ __forceinline__ void wait_async_all() {
    asm volatile("s_wait_asynccnt 0x0" ::: "memory");
}

// ============================================================================
// WMMA fp32 GEMM: C[Nr,Mout] = A[Nr,K] @ B[K,MP]   (MP = padded, mult of 16)
// One block = one 16-row tile; wave w owns column tile w. A tile + B K-chunk
// staged via global_load_async_to_lds_b128 into bank-conflict-free LDS;
// inner 16x WMMA fully unrolled.
// ============================================================================
template <int MP, int NWAVES>
__global__ void __launch_bounds__(NWAVES * 32)
wmma_gemm(const float* __restrict__ A, const float* __restrict__ B,
          float* __restrict__ C, int K, int Mout) {
    constexpr int KC   = 64;
    constexpr int MPAD = MP + 8;       // lanes16-31 (K+2 rows) land 16 banks away
    __shared__ float As[16 * 260];     // 16 rows x K (<=256), stride 260
    __shared__ float Bs[KC * MPAD];

    const int tid  = threadIdx.x;
    const int lane = tid & 31;
    const int wave = tid >> 5;
    const unsigned r0 = blockIdx.x * 16;
    const int c0   = wave * 16;

    // ---- async-stage the 16 x K A tile (16B granules) ----
    {
        const float4* Ag = (const float4*)(A + (size_t)r0 * K);
        const int k4 = K >> 2;
        for (int idx = tid; idx < 16 * k4; idx += NWAVES * 32) {
            int row = idx / k4, c4 = idx - row * k4;
            async_g2l_b128(&As[row * 260 + (c4 << 2)], Ag + idx);
        }
    }

    v8f acc = {};
    const int arow  = (lane & 15) * 260;
    const int khalf = (lane >> 4) * 2;         // lanes 16-31 hold K+2,K+3
    const int colB  = c0 + (lane & 15);

    for (int k0 = 0; k0 < K; k0 += KC) {
        // ---- async-stage B[k0..k0+KC) x MP chunk ----
        const float4* Bg = (const float4*)(B + (size_t)k0 * MP);
        constexpr int m4 = MP >> 2;
        for (int idx = tid; idx < KC * m4; idx += NWAVES * 32) {
            int row = idx / m4, c4 = idx - row * m4;
            async_g2l_b128(&Bs[row * MPAD + (c4 << 2)], Bg + idx);
        }
        if (k0 + KC < K)  // hint next chunk toward L2 (global_prefetch_b8)
            __builtin_prefetch(B + (size_t)(k0 + KC) * MP + tid, 0, 3);
        wait_async_all();      // this wave's LDS writes complete
        __syncthreads();       // all waves' writes visible

#pragma unroll
        for (int kk = 0; kk < KC; kk += 4) {
            v2f a, b;
            const int ak = k0 + kk + khalf;
            a.x = As[arow + ak];
            a.y = As[arow + ak + 1];
            const int bk = (kk + khalf) * MPAD + colB;
            b.x = Bs[bk];
            b.y = Bs[bk + MPAD];
            acc = __builtin_amdgcn_wmma_f32_16x16x4_f32(
                      false, a, false, b, (short)0, acc, false, false);
        }
        __syncthreads();       // done reading Bs before next chunk overwrites
    }

    if (colB < Mout) {
        const unsigned rbase = r0 + ((lane >> 4) ? 8u : 0u);
#pragma unroll
        for (int v = 0; v < 8; ++v)
            C[(size_t)(rbase + v) * Mout + colB] = acc[v];
    }
}

// ============================================================================
// zero-pad W2 [256,40] -> [256,48]
// ============================================================================
__global__ void pad_w2(const float* __restrict__ W2, float* __restrict__ W2p) {
    int i = blockIdx.x * blockDim.x + threadIdx.x;
    if (i >= 256 * NCLASS_P) return;
    int k = i / NCLASS_P, c = i - k * NCLASS_P;
    W2p[i] = (c < NCLASS) ? W2[k * NCLASS + c] : 0.f;
}

// ============================================================================
// Per-node attention logits: a_src[n,h] = sum_c H[n,h,c]*att_src[h,c]
// ============================================================================
__global__ void att_reduce(const float* __restrict__ H,
                           const float* __restrict__ att_src,
                           const float* __restrict__ att_dst,
                           float* __restrict__ a_src, float* __restrict__ a_dst,
                           int n, int heads, int C) {
    int i = blockIdx.x * blockDim.x + threadIdx.x;
    if (i >= n * heads) return;
    int node = i / heads, h = i - node * heads;
    const float* hp = H + (unsigned)node * (unsigned)(heads * C) + (unsigned)(h * C);
    const float* as = att_src + h * C;
    const float* ad = att_dst + h * C;
    float s1 = 0.f, s2 = 0.f;
    for (int c = 0; c < C; ++c) { float v = hp[c]; s1 += v * as[c]; s2 += v * ad[c]; }
    a_src[i] = s1; a_dst[i] = s2;
}

// ---- init helpers (re-run every launch: graph-replay safe) ----
__global__ void fill_f32(float* p, unsigned n, float v) {
    unsigned i = blockIdx.x * blockDim.x + threadIdx.x;
    if (i < n) p[i] = v;
}
__global__ void fill_u32(unsigned* p, unsigned n, unsigned v) {
    unsigned i = blockIdx.x * blockDim.x + threadIdx.x;
    if (i < n) p[i] = v;
}
__global__ void init_out_bias(float* out, const float* bias, int n, int C) {
    int i = blockIdx.x * blockDim.x + threadIdx.x;
    if (i < n * C) out[i] = bias[i % C];
}

// ============================================================================
// Edge softmax pass 1: segment_max via ordered-key atomicMax
// ============================================================================
__global__ void edge_max(const long long* __restrict__ ei,
                         const float* __restrict__ a_src,
                         const float* __restrict__ a_dst,
                         unsigned* __restrict__ emaxKey, int heads) {
    int e = blockIdx.x * blockDim.x + threadIdx.x;
    if (e >= ETOT) return;
    unsigned s, d;
    if (e < NEDGE) { s = (unsigned)ei[e]; d = (unsigned)ei[NEDGE + e]; }
    else           { s = d = (unsigned)(e - NEDGE); }
    for (int h = 0; h < heads; ++h) {
        float v = lrelu(a_src[s * heads + h] + a_dst[d * heads + h]);
        atomicMax(&emaxKey[d * heads + h], fkey(v));
    }
}

// ============================================================================
// Edge softmax pass 2: e = exp(v - max); store per-edge, accumulate denom
// ============================================================================
__global__ void edge_exp(const long long* __restrict__ ei,
                         const float* __restrict__ a_src,
                         const float* __restrict__ a_dst,
                         const unsigned* __restrict__ emaxKey,
                         float* __restrict__ edge_e,
                         float* __restrict__ denom, int heads) {
    int e = blockIdx.x * blockDim.x + threadIdx.x;
    if (e >= ETOT) return;
    unsigned s, d;
    if (e < NEDGE) { s = (unsigned)ei[e]; d = (unsigned)ei[NEDGE + e]; }
    else           { s = d = (unsigned)(e - NEDGE); }
    for (int h = 0; h < heads; ++h) {
        float v  = lrelu(a_src[s * heads + h] + a_dst[d * heads + h]);
        float m  = funkey(emaxKey[d * heads + h]);
        float ex = __expf(v - m);
        edge_e[(unsigned)e * heads + h] = ex;
        atomicAdd(&denom[d * heads + h], ex);
    }
}

// ============================================================================
// Edge message pass: one block per edge, thread t = (h,c) channel.
// out[d, h*C+c] += H[s, h*C+c] * alpha(e,h)
// ============================================================================
__global__ void edge_msg(const long long* __restrict__ ei,
                         const float* __restrict__ H,
                         const float* __restrict__ edge_e,
                         const float* __restrict__ denom,
                         float* __restrict__ out, int heads, int C) {
    int e = blockIdx.x;
    int t = threadIdx.x;
    if (t >= heads * C) return;
    unsigned s, d;
    if (e < NEDGE) { s = (unsigned)ei[e]; d = (unsigned)ei[NEDGE + e]; }
    else           { s = d = (unsigned)(e - NEDGE); }
    int h = t / C;
    float alpha = edge_e[(unsigned)e * heads + h] / (denom[d * heads + h] + 1e-16f);
    unsigned F = (unsigned)(heads * C);
    atomicAdd(&out[d * F + t], H[s * F + t] * alpha);
}

// ============================================================================
// bias + BatchNorm(eval) + ELU, in place
// ============================================================================
__global__ void bias_bn_elu(float* __restrict__ h,
                            const float* __restrict__ bias,
                            const float* __restrict__ gamma,
                            const float* __restrict__ beta,
                            const float* __restrict__ mean,
                            const float* __restrict__ var, int n, int C) {
    int i = blockIdx.x * blockDim.x + threadIdx.x;
    if (i >= n * C) return;
    int c = i % C;
    float x = h[i] + bias[c];
    x = (x - mean[c]) * rsqrtf(var[c] + BN_EPS) * gamma[c] + beta[c];
    h[i] = x > 0.f ? x : (__expf(x) - 1.f);
}

// ============================================================================
extern "C" void kernel_launch(void* const* d_in, const int* in_sizes, int n_in,
                              void* d_out, int out_size, void* d_ws, size_t ws_size,
                              hipStream_t stream) {
    const float*     x        = (const float*)d_in[0];
    const long long* ei       = (const long long*)d_in[1];   // int64 [2,E]
    const float*     W1       = (const float*)d_in[2];
    const float*     att_src1 = (const float*)d_in[3];
    const float*     att_dst1 = (const float*)d_in[4];
    const float*     bias1    = (const float*)d_in[5];
    const float*     gamma1   = (const float*)d_in[6];
    const float*     beta1    = (const float*)d_in[7];
    const float*     mean1    = (const float*)d_in[8];
    const float*     var1     = (const float*)d_in[9];
    const float*     W2       = (const float*)d_in[10];
    const float*     att_src2 = (const float*)d_in[11];
    const float*     att_dst2 = (const float*)d_in[12];
    const float*     bias2    = (const float*)d_in[13];
    float* out = (float*)d_out;

    const int FW = HEADS * HID; // 256

    // ---- workspace carve-up (floats) ----
    float* ws = (float*)d_ws;
    size_t off = 0;
    float*    H1     = ws + off; off += (size_t)NODES * FW;     // layer1 features (reused: layer2 GEMM out)
    float*    ACC    = ws + off; off += (size_t)NODES * FW;     // layer1 aggregate -> h2
    float*    aS1    = ws + off; off += (size_t)NODES * HEADS;
    float*    aD1    = ws + off; off += (size_t)NODES * HEADS;
    unsigned* eK1    = (unsigned*)(ws + off); off += (size_t)NODES * HEADS;
    float*    den1   = ws + off; off += (size_t)NODES * HEADS;
    float*    edgeE  = ws + off; off += (size_t)ETOT * HEADS;   // reused for layer2 (needs ETOT*1)
    float*    aS2    = ws + off; off += (size_t)NODES;
    float*    aD2    = ws + off; off += (size_t)NODES;
    unsigned* eK2    = (unsigned*)(ws + off); off += (size_t)NODES;
    float*    den2   = ws + off; off += (size_t)NODES;
    float*    W2p    = ws + off; off += (size_t)256 * NCLASS_P; // zero-padded W2
    (void)ws_size; (void)in_sizes; (void)n_in; (void)out_size;

    const int TB = 256;
    const int rowTiles = NODES / 16;                     // 3125, exact
    const int edgeBlocks = (ETOT + TB - 1) / TB;
    auto g = [&](size_t n) { return (unsigned)((n + TB - 1) / TB); };

    // ================= Layer 1 =================
    fill_u32<<<g((size_t)NODES * HEADS), TB, 0, stream>>>(eK1, NODES * HEADS, KEY_NEGINF);
    fill_f32<<<g((size_t)NODES * HEADS), TB, 0, stream>>>(den1, NODES * HEADS, 0.f);
    fill_f32<<<g((size_t)NODES * FW),    TB, 0, stream>>>(ACC,  NODES * FW,    0.f);
    pad_w2<<<g(256 * NCLASS_P), TB, 0, stream>>>(W2, W2p);

    // H1 = x @ W1   (256 -> 256), 16 waves: one col tile each
    wmma_gemm<256, 16><<<rowTiles, 16 * 32, 0, stream>>>(x, W1, H1, F_IN, FW);

    att_reduce<<<g((size_t)NODES * HEADS), TB, 0, stream>>>(H1, att_src1, att_dst1,
                                                            aS1, aD1, NODES, HEADS, HID);
    edge_max<<<edgeBlocks, TB, 0, stream>>>(ei, aS1, aD1, eK1, HEADS);
    edge_exp<<<edgeBlocks, TB, 0, stream>>>(ei, aS1, aD1, eK1, edgeE, den1, HEADS);
    edge_msg<<<ETOT, HEADS * HID, 0, stream>>>(ei, H1, edgeE, den1, ACC, HEADS, HID);

    bias_bn_elu<<<g((size_t)NODES * FW), TB, 0, stream>>>(ACC, bias1, gamma1, beta1,
                                                          mean1, var1, NODES, FW);

    // ================= Layer 2 =================
    float* H2P = H1;                                  // reuse: [NODES, NCLASS]
    fill_u32<<<g((size_t)NODES), TB, 0, stream>>>(eK2, NODES, KEY_NEGINF);
    fill_f32<<<g((size_t)NODES), TB, 0, stream>>>(den2, NODES, 0.f);
    init_out_bias<<<g((size_t)NODES * NCLASS), TB, 0, stream>>>(out, bias2, NODES, NCLASS);

    // H2P = h2 @ W2p  (256 -> 48 padded, store 40), 3 col tiles -> 3 waves
    wmma_gemm<NCLASS_P, 3><<<rowTiles, 3 * 32, 0, stream>>>(ACC, W2p, H2P, FW, NCLASS);

    att_reduce<<<g((size_t)NODES), TB, 0, stream>>>(H2P, att_src2, att_dst2,
                                                    aS2, aD2, NODES, 1, NCLASS);
    edge_max<<<edgeBlocks, TB, 0, stream>>>(ei, aS2, aD2, eK2, 1);
    edge_exp<<<edgeBlocks, TB, 0, stream>>>(ei, aS2, aD2, eK2, edgeE, den2, 1);
    edge_msg<<<ETOT, 64, 0, stream>>>(ei, H2P, edgeE, den2, out, 1, NCLASS);
}